// Remesher_28544352649766
// MI455X (gfx1250) — compile-verified
//
#include <hip/hip_runtime.h>
#include <stdint.h>

typedef unsigned long long u64;
typedef uint32_t u32;

// Strictly monotone map from (priority, edge_index) to u64, matching the
// ordering of rank = argsort(priorities) (stable ties by index). Max-propagation
// over these keys is exactly equivalent to max-propagation over ranks.
__device__ __forceinline__ u64 edge_key0(float p, int e) {
    u32 u = __float_as_uint(p);
    u = (u & 0x80000000u) ? ~u : (u | 0x80000000u);   // total order on floats
    return ((u64)u << 32) | (u32)e;
}

// ---------------- init: vert_key=0, vert_conn=0, dest=identity ----------------
__global__ void __launch_bounds__(256) k_init(u64* vert_key, int* vert_conn, int* dest, int V) {
    int i = blockIdx.x * blockDim.x + threadIdx.x;
    if (i < V) vert_key[i] = 0ull;
    if (i <= V) { vert_conn[i] = 0; dest[i] = i; }
}

// ---------------- rank (key) propagation ----------------
__global__ void __launch_bounds__(256) k_scatter0(const int2* __restrict__ edges,
                                                  const float* __restrict__ prio,
                                                  u64* vert_key, int E) {
    int e = blockIdx.x * blockDim.x + threadIdx.x;
    if (e >= E) return;
    int2 ed = edges[e];
    u64 k = edge_key0(prio[e], e);
    atomicMax(&vert_key[ed.x], k);
    atomicMax(&vert_key[ed.y], k);
}

__global__ void __launch_bounds__(256) k_scatter(const int2* __restrict__ edges,
                                                 const u64* __restrict__ edge_key,
                                                 u64* vert_key, int E) {
    int e = blockIdx.x * blockDim.x + threadIdx.x;
    if (e >= E) return;
    int2 ed = edges[e];
    u64 k = edge_key[e];
    atomicMax(&vert_key[ed.x], k);
    atomicMax(&vert_key[ed.y], k);
}

// Gather: stage the contiguous edge stream into LDS via the CDNA5 async
// global->LDS copy (ASYNCcnt-tracked), then do the random vert_key gathers.
__global__ void __launch_bounds__(256) k_gather(const int2* __restrict__ edges,
                                                const u64* __restrict__ vert_key,
                                                u64* __restrict__ edge_key, int E) {
    __shared__ int2 stage[256];
    int e = blockIdx.x * blockDim.x + threadIdx.x;
    if (e < E) {
        // Low 32 bits of a generic pointer to LDS are the 0-based LDS byte offset
        // (ISA 10.2).
        u32 lds_off = (u32)(uintptr_t)&stage[threadIdx.x];
        const int2* gp = &edges[e];
        asm volatile("global_load_async_to_lds_b64 %0, %1, off"
                     :: "v"(lds_off), "v"(gp) : "memory");
        asm volatile("s_wait_asynccnt 0" ::: "memory");
        int2 ed = stage[threadIdx.x];               // each lane reads its own slot
        u64 a = vert_key[ed.x];
        u64 b = vert_key[ed.y];
        edge_key[e] = a > b ? a : b;
    }
}

// Final gather fused with candidate test AND the first connectivity mark:
//   cand = (propagated key == own key) & (prio > 0)
//   if (cand) vert_conn[e0] = 1      (reference: vert_conn.at[c0].set(1))
// The mark depends only on this edge's own cand value -> safe to fuse;
// candidate edges form a matching -> plain store is race-free.
__global__ void __launch_bounds__(256) k_gather_cand(const int2* __restrict__ edges,
                                                     const float* __restrict__ prio,
                                                     const u64* __restrict__ vert_key,
                                                     unsigned char* __restrict__ cand,
                                                     int* vert_conn, int E) {
    __shared__ int2 stage[256];
    int e = blockIdx.x * blockDim.x + threadIdx.x;
    if (e < E) {
        u32 lds_off = (u32)(uintptr_t)&stage[threadIdx.x];
        const int2* gp = &edges[e];
        asm volatile("global_load_async_to_lds_b64 %0, %1, off"
                     :: "v"(lds_off), "v"(gp) : "memory");
        asm volatile("s_wait_asynccnt 0" ::: "memory");
        int2 ed = stage[threadIdx.x];
        u64 a = vert_key[ed.x];
        u64 b = vert_key[ed.y];
        u64 m = a > b ? a : b;
        float p = prio[e];
        bool c = (m == edge_key0(p, e)) && (p > 0.0f);
        cand[e] = c ? 1 : 0;
        if (c) vert_conn[ed.x] = 1;
    }
}

// ---------------- connectivity counting ----------------
__global__ void __launch_bounds__(256) k_conn_gather(const int2* __restrict__ edges,
                                                     const int* __restrict__ vert_conn,
                                                     int* __restrict__ edge_conn, int E) {
    int e = blockIdx.x * blockDim.x + threadIdx.x;
    if (e >= E) return;
    int2 ed = edges[e];
    edge_conn[e] = vert_conn[ed.x] + vert_conn[ed.y];
}

__global__ void __launch_bounds__(256) k_conn_add(const int2* __restrict__ edges,
                                                  const int* __restrict__ edge_conn,
                                                  int* vert_conn, int E) {
    int e = blockIdx.x * blockDim.x + threadIdx.x;
    if (e >= E) return;
    int2 ed = edges[e];
    int c = edge_conn[e];
    atomicAdd(&vert_conn[ed.x], c);
    atomicAdd(&vert_conn[ed.y], c);
}

__global__ void __launch_bounds__(256) k_conn_zero(const int2* __restrict__ edges,
                                                   const unsigned char* __restrict__ cand,
                                                   int* vert_conn, int E) {
    int e = blockIdx.x * blockDim.x + threadIdx.x;
    if (e >= E || !cand[e]) return;
    int2 ed = edges[e];
    vert_conn[ed.x] = 0;
    vert_conn[ed.y] = 0;
}

// ---------------- outputs ----------------
__global__ void __launch_bounds__(256) k_copy_verts(const float4* __restrict__ src4,
                                                    float4* __restrict__ dst4,
                                                    const float* __restrict__ src,
                                                    float* __restrict__ dst,
                                                    int n4, int n) {
    int i = blockIdx.x * blockDim.x + threadIdx.x;
    if (i < n4) dst4[i] = src4[i];                  // b128 bulk copy
    int t = 4 * n4 + i;
    if (t < n) dst[t] = src[t];                     // scalar tail
}

// Fused finalize over edges:
//   collapse = cand & (vert_conn[e1] <= 2)   (vert_conn after 2nd add round)
//   out_v[e0] = 0.5*(v[e0]+v[e1])            (reference: vertices.at[s0].set(mid))
//   dest[e1]  = e0                           (reference: dest.at[s1].set(e0))
// Collapse edges are a matching -> all stores race-free.
__global__ void __launch_bounds__(256) k_finalize(const int2* __restrict__ edges,
                                                  const unsigned char* __restrict__ cand,
                                                  const int* __restrict__ vert_conn,
                                                  const float* __restrict__ verts,
                                                  float* __restrict__ out_v,
                                                  int* __restrict__ dest, int E) {
    int e = blockIdx.x * blockDim.x + threadIdx.x;
    if (e >= E || !cand[e]) return;
    int2 ed = edges[e];
    if (vert_conn[ed.y] > 2) return;
    const float* a = verts + (size_t)3 * ed.x;
    const float* b = verts + (size_t)3 * ed.y;
    float* o = out_v + (size_t)3 * ed.x;
    o[0] = 0.5f * (a[0] + b[0]);
    o[1] = 0.5f * (a[1] + b[1]);
    o[2] = 0.5f * (a[2] + b[2]);
    dest[ed.y] = ed.x;
}

__global__ void __launch_bounds__(256) k_faces(const int* __restrict__ faces,
                                               const int* __restrict__ dest,
                                               int* __restrict__ out_f, int F) {
    int f = blockIdx.x * blockDim.x + threadIdx.x;
    if (f >= F) return;
    size_t base = (size_t)3 * f;
    // faces stream is touched exactly once -> non-temporal; dest (8MB) stays hot in L2.
    int i0 = __builtin_nontemporal_load(&faces[base + 0]);
    int i1 = __builtin_nontemporal_load(&faces[base + 1]);
    int i2 = __builtin_nontemporal_load(&faces[base + 2]);
    int a = dest[i0];
    int b = dest[i1];
    int c = dest[i2];
    bool collapsed = (a == b) | (b == c) | (a == c);
    __builtin_nontemporal_store(collapsed ? 0 : a, &out_f[base + 0]);
    __builtin_nontemporal_store(collapsed ? 0 : b, &out_f[base + 1]);
    __builtin_nontemporal_store(collapsed ? 0 : c, &out_f[base + 2]);
}

extern "C" void kernel_launch(void* const* d_in, const int* in_sizes, int n_in,
                              void* d_out, int out_size, void* d_ws, size_t ws_size,
                              hipStream_t stream) {
    const float* vertices = (const float*)d_in[0];
    const int*   faces    = (const int*)d_in[1];
    const int2*  edges    = (const int2*)d_in[2];
    const float* prio     = (const float*)d_in[3];
    const int V = in_sizes[0] / 3;
    const int F = in_sizes[1] / 3;
    const int E = in_sizes[3];

    // Workspace carve-out (~110 MB total), 256B aligned slices.
    char* w = (char*)d_ws;
    auto take = [&](size_t bytes) -> void* {
        void* p = (void*)w;
        w += (bytes + 255) & ~(size_t)255;
        return p;
    };
    u64* vert_key       = (u64*)take(sizeof(u64) * (size_t)V);
    u64* edge_key       = (u64*)take(sizeof(u64) * (size_t)E);
    int* vert_conn      = (int*)take(sizeof(int) * (size_t)(V + 1));
    int* edge_conn      = (int*)take(sizeof(int) * (size_t)E);
    int* dest           = (int*)take(sizeof(int) * (size_t)(V + 1));
    unsigned char* cand = (unsigned char*)take((size_t)E);

    float* out_v = (float*)d_out;
    int*   out_f = (int*)d_out + (size_t)3 * V;

    const int B = 256;
    auto g = [&](long long n) { return (unsigned)((n + B - 1) / B); };

    k_init<<<g(V + 1), B, 0, stream>>>(vert_key, vert_conn, dest, V);

    // 3 rounds of scatter-max / gather-max over (priority,index) keys == rank propagation
    k_scatter0<<<g(E), B, 0, stream>>>(edges, prio, vert_key, E);
    k_gather<<<g(E), B, 0, stream>>>(edges, vert_key, edge_key, E);
    k_scatter<<<g(E), B, 0, stream>>>(edges, edge_key, vert_key, E);
    k_gather<<<g(E), B, 0, stream>>>(edges, vert_key, edge_key, E);
    k_scatter<<<g(E), B, 0, stream>>>(edges, edge_key, vert_key, E);
    k_gather_cand<<<g(E), B, 0, stream>>>(edges, prio, vert_key, cand, vert_conn, E);

    // connectivity counting (two rounds; mark-1 already fused into gather_cand)
    k_conn_gather<<<g(E), B, 0, stream>>>(edges, vert_conn, edge_conn, E);
    k_conn_add<<<g(E), B, 0, stream>>>(edges, edge_conn, vert_conn, E);
    k_conn_zero<<<g(E), B, 0, stream>>>(edges, cand, vert_conn, E);
    k_conn_gather<<<g(E), B, 0, stream>>>(edges, vert_conn, edge_conn, E);
    k_conn_add<<<g(E), B, 0, stream>>>(edges, edge_conn, vert_conn, E);

    // outputs: bulk vertex copy, then fused collapse (mid-scatter + dest map), then faces
    int n = 3 * V, n4 = n / 4;
    k_copy_verts<<<g(n4 + 4), B, 0, stream>>>((const float4*)vertices, (float4*)out_v,
                                              vertices, out_v, n4, n);
    k_finalize<<<g(E), B, 0, stream>>>(edges, cand, vert_conn, vertices, out_v, dest, E);
    k_faces<<<g(F), B, 0, stream>>>(faces, dest, out_f, F);
}